// ScaledTurnEmbedding_65781719106240
// MI455X (gfx1250) — compile-verified
//
#include <hip/hip_runtime.h>

typedef __attribute__((ext_vector_type(2))) float v2f;
typedef __attribute__((ext_vector_type(4))) float v4f;
typedef __attribute__((ext_vector_type(8))) float v8f;

#define N_TURNS    8
#define OUT_DIM    768
#define S_LEN      2048
#define LDS_STRIDE 772      // dwords; 3088 B per row: 16B-multiple for b128, 8*772 % 64 == 32
                            // so the two half-wave bank windows stay disjoint on the WMMA store

__global__ __launch_bounds__(256) void ScaledTurnEmbedding_wmma_kernel(
    const float* __restrict__ turns,      // [VOCAB, 8]
    const float* __restrict__ coeffs,     // [8, 5, 768] == row-major [40, 768]
    const float* __restrict__ pos_table,  // [2048, 768]
    const float* __restrict__ gamma,      // [768]
    const float* __restrict__ beta,       // [768]
    const int*   __restrict__ ids,        // [16, 2048] flat
    float* __restrict__ out)              // [32768, 768]
{
    __shared__ __align__(16) float lds_tile[16 * LDS_STRIDE];
    __shared__ __align__(16) v2f   red[16 * 16];   // {sum, sumsq} partials

    const int tid  = threadIdx.x;
    const int wave = tid >> 5;          // 0..7
    const int lane = tid & 31;
    const int l    = lane & 15;
    const int h    = lane >> 4;         // 0: K pair {0,1}, 1: K pair {2,3} within a k-step
    const int g0   = (int)blockIdx.x * 16;   // first flat token of this tile

    // ---- per-lane power table for token g0 + l (A-matrix row) ----
    const int tok_id = ids[g0 + l];
    float pw[40];
    #pragma unroll
    for (int t = 0; t < N_TURNS; ++t) {
        const float x  = turns[tok_id * N_TURNS + t];
        const float x2 = x * x;
        pw[t*5 + 0] = 1.0f;
        pw[t*5 + 1] = x;
        pw[t*5 + 2] = x2;
        pw[t*5 + 3] = x2 * x;
        pw[t*5 + 4] = x2 * x2;
    }
    // A fragments: lane (l, h) holds A[l][4*kb + 2h] and A[l][4*kb + 2h + 1]
    v2f a_frag[10];
    #pragma unroll
    for (int kb = 0; kb < 10; ++kb) {
        a_frag[kb].x = h ? pw[kb*4 + 2] : pw[kb*4 + 0];
        a_frag[kb].y = h ? pw[kb*4 + 3] : pw[kb*4 + 1];
    }

    // ---- GEMM: 6 N-tiles of 16 dims per wave, K = 40 in 10 wmma steps ----
    #pragma unroll 1
    for (int nt = 0; nt < 6; ++nt) {
        const int d = (wave * 6 + nt) * 16 + l;   // dim owned by this lane for B / C / D
        v8f c = {0.f, 0.f, 0.f, 0.f, 0.f, 0.f, 0.f, 0.f};
        #pragma unroll
        for (int kb = 0; kb < 10; ++kb) {
            const int kk = kb * 4 + 2 * h;        // B rows (k, k+1) for this lane half
            v2f b;
            b.x = coeffs[ kk      * OUT_DIM + d];
            b.y = coeffs[(kk + 1) * OUT_DIM + d];
            // D = A x B + C : emb[m][d] += sum_k powers[m][k] * coeffs[k][d]
            c = __builtin_amdgcn_wmma_f32_16x16x4_f32(
                    false, a_frag[kb], false, b, (short)0, c, false, false);
        }
        #pragma unroll
        for (int r = 0; r < 8; ++r) {
            const int m = r + 8 * h;              // token row within the 16-token tile
            lds_tile[m * LDS_STRIDE + d] = c[r];
        }
    }
    __syncthreads();

    // ---- fused pos-add + LayerNorm: 16 threads per token row, float4-wide ----
    const int tk = tid >> 4;                      // token within tile, 0..15
    const int j  = tid & 15;
    const int s  = (g0 & (S_LEN - 1)) + tk;       // sequence position (tiles never cross rows)

    const float* __restrict__ prow = pos_table + (size_t)s * OUT_DIM;
    v4f vals[12];
    float sum = 0.f, sq = 0.f;
    #pragma unroll
    for (int i = 0; i < 12; ++i) {
        const int d4 = (j + 16 * i) * 4;          // 16B-aligned chunk of 4 dims
        const v4f e = *(const v4f*)(&lds_tile[tk * LDS_STRIDE + d4]);   // ds_load_b128
        const v4f p = *(const v4f*)(&prow[d4]);                          // global_load_b128
        const v4f v = e + p;
        vals[i] = v;
        sum += v.x + v.y + v.z + v.w;
        sq  += v.x * v.x + v.y * v.y + v.z * v.z + v.w * v.w;
    }
    red[tk * 16 + j] = (v2f){sum, sq};            // ds_store_b64
    __syncthreads();

    v2f tot = (v2f){0.f, 0.f};
    #pragma unroll
    for (int i = 0; i < 16; ++i)
        tot += red[tk * 16 + i];
    const float mu  = tot.x * (1.0f / OUT_DIM);
    const float var = tot.y * (1.0f / OUT_DIM) - mu * mu;   // E[x^2] - mu^2, matches reference
    const float rs  = rsqrtf(var + 1e-12f);

    float* __restrict__ orow = out + (size_t)(g0 + tk) * OUT_DIM;
    #pragma unroll
    for (int i = 0; i < 12; ++i) {
        const int d4 = (j + 16 * i) * 4;
        const v4f g = *(const v4f*)(&gamma[d4]);
        const v4f b = *(const v4f*)(&beta[d4]);
        const v4f o = (vals[i] - mu) * rs * g + b;
        *(v4f*)(&orow[d4]) = o;                   // global_store_b128, 256B/group coalesced
    }
}

extern "C" void kernel_launch(void* const* d_in, const int* in_sizes, int n_in,
                              void* d_out, int out_size, void* d_ws, size_t ws_size,
                              hipStream_t stream) {
    const float* turns     = (const float*)d_in[0];
    const float* coeffs    = (const float*)d_in[1];
    const float* pos_table = (const float*)d_in[2];
    const float* gamma     = (const float*)d_in[3];
    const float* beta      = (const float*)d_in[4];
    const int*   ids       = (const int*)  d_in[5];
    float*       out       = (float*)d_out;

    const int n_tokens = in_sizes[5];             // B * S = 32768
    const int n_blocks = n_tokens / 16;           // 2048 workgroups, one 16-token tile each

    ScaledTurnEmbedding_wmma_kernel<<<n_blocks, 256, 0, stream>>>(
        turns, coeffs, pos_table, gamma, beta, ids, out);
}